// OptimizedLowRankRNN_58145267253496
// MI455X (gfx1250) — compile-verified
//
#include <hip/hip_runtime.h>

// Problem constants (from reference): B=128, T=1000, I=3, H=512, O=1, R=2
// ALPHA = 0.2, NOISE_STD = 0.05

typedef float v2f __attribute__((ext_vector_type(2)));
typedef float v8f __attribute__((ext_vector_type(8)));

__device__ __forceinline__ float fast_tanh(float x) {
  // tanh(x) = sign(x) * (1 - e) / (1 + e), e = exp(-2|x|) = exp2(-2*log2e*|x|)
  float ax = __builtin_fabsf(x);
  float e  = __builtin_amdgcn_exp2f(-2.8853900817779268f * ax);
  float r  = (1.0f - e) * __builtin_amdgcn_rcpf(1.0f + e);
  return __builtin_copysignf(r, x);
}

__device__ __forceinline__ float wave_sum(float v) {
  v += __shfl_xor(v, 16, 32);
  v += __shfl_xor(v, 8, 32);
  v += __shfl_xor(v, 4, 32);
  v += __shfl_xor(v, 2, 32);
  v += __shfl_xor(v, 1, 32);
  return v;
}

// ---------------------------------------------------------------------------
// Kernel 1: time-parallel drive precompute via WMMA.
//   drive[b,t,:] = 0.05*noise[b,t,:] + 0.2*(x[b,t,:] @ (si[:,None]*wi))
// computed as D = A x B + C with V_WMMA_F32_16X16X4_F32:
//   A[M,k] = 0.2*si[k]*x[row,k]  (16 bt-rows x K=4, K=3 used)
//   B[k,n] = wi[k, hb+n]         (4 x 16 tile, row k=3 zero)
//   C[M,n] = 0.05*noise[row, hb+n]
// Result written into trajectory rows 1..T of d_out (later overwritten by h).
// All lane-dependent operand construction is branch-free (cndmask, not
// EXEC-masked loads) so EXEC stays all-1s for the WMMA with no per-iteration
// save/restore-exec traffic.
// ---------------------------------------------------------------------------
__global__ __launch_bounds__(256) void lrrnn_drive_wmma(
    const float* __restrict__ input,   // (B*T, 3)
    const float* __restrict__ noise,   // (B*T, 512)
    const float* __restrict__ wi,      // (3, 512)
    const float* __restrict__ si,      // (3,)
    float* __restrict__ traj) {        // (B, 1001, 512)
  const int  lane   = threadIdx.x & 31;
  const int  rt     = blockIdx.x * 8 + (threadIdx.x >> 5);  // row-tile 0..7999
  const int  half   = lane >> 4;   // 0: lanes 0-15, 1: lanes 16-31
  const int  l15    = lane & 15;
  const bool hi     = (half != 0);

  // A-matrix layout (16x4 f32): lanes 0-15 hold (M=lane, K=0..1) in v0..v1,
  // lanes 16-31 hold (M=lane-16, K=2..3).  Branch-free construction.
  v2f A;
  {
    const float* xr = input + (size_t)(rt * 16 + l15) * 3;
    const float x0 = xr[0], x1 = xr[1], x2 = xr[2];
    const float s0 = si[0], s1 = si[1], s2 = si[2];
    A[0] = 0.2f * (hi ? s2 * x2 : s0 * x0);
    A[1] = hi ? 0.0f : 0.2f * s1 * x1;     // K=3 pad on upper half
  }

  // C/D layout (16x16 f32): VGPR v, lane -> (M = v + half*8, N = l15).
  const float* nrow[8];
  float*       drow[8];
#pragma unroll
  for (int v = 0; v < 8; ++v) {
    const int row = rt * 16 + half * 8 + v;    // bt row
    const int bb  = row / 1000;
    const int tt  = row - bb * 1000;
    nrow[v] = noise + (size_t)row * 512 + l15;
    drow[v] = traj + ((size_t)bb * 1001 + (size_t)(tt + 1)) * 512 + l15;
  }

  // B-matrix (4x16): v0 -> (K = 2*half, N = l15), v1 -> (K = 2*half+1, N = l15)
  // K=3 row is zero.  Load unconditionally (always in-bounds: wi has 1536
  // elements and 512+hb+l15 <= 1023), select pad lanes to zero afterwards.
  const float* wiA = wi + (hi ? 2 * 512 : 0) + l15;  // K=0 or K=2
  const float* wiB = wi + 512 + l15;                 // K=1 (or dummy for K=3)

#pragma unroll 4
  for (int ht = 0; ht < 32; ++ht) {
    const int hb = ht * 16;
    const float b1raw = wiB[hb];           // unconditional, branch-free
    v2f Bv;
    Bv[0] = wiA[hb];
    Bv[1] = hi ? 0.0f : b1raw;             // v_cndmask, no exec masking
    v8f C;
#pragma unroll
    for (int v = 0; v < 8; ++v) C[v] = 0.05f * nrow[v][hb];
    v8f D = __builtin_amdgcn_wmma_f32_16x16x4_f32(
        /*neg_a=*/false, A, /*neg_b=*/false, Bv,
        /*c_mod=*/(short)0, C, /*reuse_a=*/false, /*reuse_b=*/false);
#pragma unroll
    for (int v = 0; v < 8; ++v) drow[v][hb] = D[v];
  }
}

// ---------------------------------------------------------------------------
// Kernel 2: sequential scan, one batch element per wave32 (no barriers).
// Lane owns H-elements [lane*16, lane*16+16). Per step:
//   r = tanh(h); p = r@n (intra-wave reduce); h += drive + 0.2*(m@p - h);
//   out[b,t] = tanh(h_new)@wo_full  (reuses next step's tanh values).
// ---------------------------------------------------------------------------
__global__ __launch_bounds__(32) void lrrnn_scan(
    const float* __restrict__ m,    // (512, 2)
    const float* __restrict__ n,    // (512, 2)
    const float* __restrict__ wo,   // (512, 1)
    const float* __restrict__ so,   // (1,)
    const float* __restrict__ h0,   // (512,)
    float* __restrict__ out,        // (B, T)
    float* __restrict__ traj) {     // (B, 1001, 512); rows 1..T hold drive
  const int b    = blockIdx.x;
  const int lane = threadIdx.x;
  const int g0   = lane * 16;

  float h[16], m0[16], m1[16], n0[16], n1[16], wf[16];
  const float so0 = so[0];
#pragma unroll
  for (int j = 0; j < 16; ++j) {
    const int g = g0 + j;
    n0[j] = n[2 * g];
    n1[j] = n[2 * g + 1];
    m0[j] = m[2 * g];
    m1[j] = m[2 * g + 1];
    wf[j] = wo[g] * so0;
    h[j]  = h0[g];
  }

  float* trow = traj + (size_t)b * 1001 * 512 + g0;

  // trajectories[b, 0, :] = h0
#pragma unroll
  for (int q = 0; q < 4; ++q)
    *(float4*)(trow + 4 * q) =
        make_float4(h[4 * q], h[4 * q + 1], h[4 * q + 2], h[4 * q + 3]);

  for (int t = 0; t < 1000; ++t) {
    float* cur = trow + (size_t)(t + 1) * 512;
    // Load drive for this step (overlaps with tanh chain below).
    float4 d0 = *(const float4*)(cur + 0);
    float4 d1 = *(const float4*)(cur + 4);
    float4 d2 = *(const float4*)(cur + 8);
    float4 d3 = *(const float4*)(cur + 12);
    if (t + 2 <= 1000)
      __builtin_prefetch(trow + (size_t)(t + 2) * 512, 0, 1);

    float p0 = 0.f, p1 = 0.f, oacc = 0.f;
#pragma unroll
    for (int j = 0; j < 16; ++j) {
      const float r = fast_tanh(h[j]);   // tanh of h_new from step t-1
      p0   += r * n0[j];
      p1   += r * n1[j];
      oacc += r * wf[j];
    }
    p0   = wave_sum(p0);
    p1   = wave_sum(p1);
    oacc = wave_sum(oacc);
    if (t > 0 && lane == 0) out[b * 1000 + (t - 1)] = oacc;

    const float dd[16] = {d0.x, d0.y, d0.z, d0.w, d1.x, d1.y, d1.z, d1.w,
                          d2.x, d2.y, d2.z, d2.w, d3.x, d3.y, d3.z, d3.w};
#pragma unroll
    for (int j = 0; j < 16; ++j) {
      const float rec = m0[j] * p0 + m1[j] * p1;
      h[j] = h[j] + dd[j] + 0.2f * (rec - h[j]);
    }
    // trajectories[b, t+1, :] = h_new (overwrites the drive row)
#pragma unroll
    for (int q = 0; q < 4; ++q)
      *(float4*)(cur + 4 * q) =
          make_float4(h[4 * q], h[4 * q + 1], h[4 * q + 2], h[4 * q + 3]);
  }

  // Final output uses tanh of the last h_new.
  float of = 0.f;
#pragma unroll
  for (int j = 0; j < 16; ++j) of += fast_tanh(h[j]) * wf[j];
  of = wave_sum(of);
  if (lane == 0) out[b * 1000 + 999] = of;
}

// ---------------------------------------------------------------------------
extern "C" void kernel_launch(void* const* d_in, const int* in_sizes, int n_in,
                              void* d_out, int out_size, void* d_ws, size_t ws_size,
                              hipStream_t stream) {
  (void)in_sizes; (void)n_in; (void)out_size; (void)d_ws; (void)ws_size;
  const float* input = (const float*)d_in[0];  // (128,1000,3)
  const float* noise = (const float*)d_in[1];  // (128,1000,512)
  const float* wi    = (const float*)d_in[2];  // (3,512)
  const float* si    = (const float*)d_in[3];  // (3,)
  const float* m     = (const float*)d_in[4];  // (512,2)
  const float* nn    = (const float*)d_in[5];  // (512,2)
  const float* wo    = (const float*)d_in[6];  // (512,1)
  const float* so    = (const float*)d_in[7];  // (1,)
  const float* h0    = (const float*)d_in[8];  // (512,)

  float* out  = (float*)d_out;          // (128,1000,1) = 128000 floats
  float* traj = out + 128 * 1000;       // (128,1001,512)

  // 8000 row-tiles of 16 bt-rows; 8 waves per block -> 1000 blocks.
  lrrnn_drive_wmma<<<1000, 256, 0, stream>>>(input, noise, wi, si, traj);
  // One wave32 per batch element.
  lrrnn_scan<<<128, 32, 0, stream>>>(m, nn, wo, so, h0, out, traj);
}